// S2ConvNet_86792699118281
// MI455X (gfx1250) — compile-verified
//
#include <hip/hip_runtime.h>
#include <hip/hip_bf16.h>
#include <math.h>

typedef __attribute__((ext_vector_type(2))) float v2f;
typedef __attribute__((ext_vector_type(8))) float v8f;
typedef __attribute__((ext_vector_type(4))) unsigned int u32x4;
typedef __attribute__((ext_vector_type(8))) int i32x8;
typedef __attribute__((ext_vector_type(4))) int i32x4;

constexpr double PI_D = 3.14159265358979323846;
#define TWO_PI_F 6.28318530717958647692f

// ---- architecture constants ----
constexpr int BATCH=128, BIN=30, B1=10, B2=6, F1=20, F2=40, FOUT=10;
constexpr int NB_IN=60;   // 2*BIN
constexpr int NB1=20;     // 2*B1
constexpr int NB2=12;     // 2*B2
constexpr int M1=19;      // 2*B1-1 retained azimuthal freqs
constexpr int M2=11;      // 2*B2-1

// ---- table sizes (floats) ----
constexpr size_t SZ_WD   = (size_t)B1*NB_IN*M1;        // 11400
constexpr size_t SZ_Y1   = (size_t)B1*24*M1;           // 4560 per plane
constexpr size_t SZ_D1W  = (size_t)B1*NB1*M1*M1;       // 72200
constexpr size_t SZ_D12W = (size_t)B2*NB1*M2*M2;       // 14520
constexpr size_t SZ_D2   = (size_t)B2*144*M2*M2;       // 104544 per plane
constexpr size_t SZ_D2W  = (size_t)B2*NB2*M2*M2;       // 8712
constexpr size_t SZ_W2N  = NB2;

constexpr size_t SZ_X    = (size_t)B1*BATCH*M1;
constexpr size_t SZ_P1   = (size_t)B1*F1*M1;
constexpr size_t SZ_FEAT = (size_t)BATCH*F2;

constexpr size_t SZ_Y    = (size_t)BATCH*F1*NB1*NB1*NB1; // 20.48M floats
constexpr size_t SZ_YF   = (size_t)BATCH*F1*NB1*M2*M2;   // per plane

// GEMM dims (complex, per l): Fo[M,N] = Fx[M,K] x Psi2[K,N]
constexpr int GM=BATCH*M2;          // 1408 = 88*16
constexpr int GK=F1*M2;             // 220  (multiple of 4)
constexpr int GN=F2*M2;             // 440
constexpr int GNP=448;              // padded to 28*16
constexpr size_t SZ_FX  = (size_t)B2*GM*GK;
constexpr size_t SZ_P2  = (size_t)B2*GK*GNP;
constexpr size_t SZ_FO  = (size_t)B2*GM*GNP;

constexpr size_t AL(size_t x){ return (x+63)&~(size_t)63; }
constexpr size_t OFF_WD  = 0;
constexpr size_t OFF_Y1R = AL(OFF_WD+SZ_WD);
constexpr size_t OFF_Y1I = AL(OFF_Y1R+SZ_Y1);
constexpr size_t OFF_D1W = AL(OFF_Y1I+SZ_Y1);
constexpr size_t OFF_D12W= AL(OFF_D1W+SZ_D1W);
constexpr size_t OFF_D2R = AL(OFF_D12W+SZ_D12W);
constexpr size_t OFF_D2I = AL(OFF_D2R+SZ_D2);
constexpr size_t OFF_D2W = AL(OFF_D2I+SZ_D2);
constexpr size_t OFF_W2N = AL(OFF_D2W+SZ_D2W);
constexpr size_t OFF_XR  = AL(OFF_W2N+SZ_W2N);
constexpr size_t OFF_XI  = AL(OFF_XR+SZ_X);
constexpr size_t OFF_P1R = AL(OFF_XI+SZ_X);
constexpr size_t OFF_P1I = AL(OFF_P1R+SZ_P1);
constexpr size_t OFF_FEAT= AL(OFF_P1I+SZ_P1);
constexpr size_t OFF_Y   = AL(OFF_FEAT+SZ_FEAT);
constexpr size_t OFF_YFR = AL(OFF_Y+SZ_Y);
constexpr size_t OFF_YFI = AL(OFF_YFR+SZ_YF);
// GEMM buffers overlay the y region (y is dead once yf is built)
constexpr size_t OFF_FXR = OFF_Y;
constexpr size_t OFF_FXI = AL(OFF_FXR+SZ_FX);
constexpr size_t OFF_P2R = AL(OFF_FXI+SZ_FX);
constexpr size_t OFF_P2I = AL(OFF_P2R+SZ_P2);
constexpr size_t OFF_FOR = AL(OFF_P2I+SZ_P2);
constexpr size_t OFF_FOI = AL(OFF_FOR+SZ_FO);
static_assert(OFF_FOI+SZ_FO <= OFF_Y+SZ_Y, "GEMM overlay must fit in y region");

// ======================= constant tables (Wigner-d etc.) =======================
__device__ __forceinline__ double d_lf(int k){ return lgamma((double)k+1.0); }

__device__ double wig_d(int l,int m,int n,double beta){
  if (m>l||m<-l||n>l||n<-l) return 0.0;
  double cb=cos(0.5*beta), sb=sin(0.5*beta);
  double pref=0.5*(d_lf(l+m)+d_lf(l-m)+d_lf(l+n)+d_lf(l-n));
  int s0=(n-m>0)?(n-m):0;
  int s1=(l+n<l-m)?(l+n):(l-m);
  double acc=0.0;
  for(int s=s0;s<=s1;++s){
    double c=pref-(d_lf(l+n-s)+d_lf(s)+d_lf(m-n+s)+d_lf(l-m-s));
    double sgn=((m-n+s)&1)?-1.0:1.0;
    acc += sgn*exp(c)*pow(cb,(double)(2*l+n-m-2*s))*pow(sb,(double)(m-n+2*s));
  }
  return acc;
}
__device__ __forceinline__ double dh_beta(int b,int k){ return PI_D*(2*k+1)/(4.0*b); }
__device__ double dh_w(int b,int k){
  double beta=dh_beta(b,k), s=0.0;
  for(int j=0;j<b;++j) s+=sin((2.0*j+1.0)*beta)/(2.0*j+1.0);
  return (2.0/b)*sin(beta)*s;
}

__global__ void k_tables(float* ws){
  size_t e = blockIdx.x*(size_t)blockDim.x + threadIdx.x;
  if (e < SZ_WD){ // quadrature-weighted d^l_{m0} on b_in beta grid [10,60,19]
    int l=(int)(e/(NB_IN*M1)); int r=(int)(e%(NB_IN*M1)); int b=r/M1; int mi=r%M1; int m=mi-(B1-1);
    ws[OFF_WD+e]=(float)(dh_w(BIN,b)*wig_d(l,m,0,dh_beta(BIN,b)));
    return;
  }
  e-=SZ_WD;
  if (e < SZ_Y1){ // conj spherical harmonics on 24-pt S2 grid [10,24,19] complex
    int l=(int)(e/(24*M1)); int r=(int)(e%(24*M1)); int g=r/M1; int mi=r%M1; int m=mi-(B1-1);
    double beta=(g/8+1)*(PI_D/24.0);
    double alpha=(g%8)*(2.0*PI_D/8.0);
    double d=wig_d(l,m,0,beta), ph=m*alpha;
    ws[OFF_Y1R+e]=(float)( d*cos(ph));
    ws[OFF_Y1I+e]=(float)(-d*sin(ph));
    return;
  }
  e-=SZ_Y1;
  if (e < SZ_D1W){ // (2l+1)*d^l on b1 beta grid [10,20,19,19]
    int l=(int)(e/(NB1*M1*M1)); int r=(int)(e%(NB1*M1*M1)); int b=r/(M1*M1); int mn=r%(M1*M1);
    int m=mn/M1-(B1-1), n=mn%M1-(B1-1);
    ws[OFF_D1W+e]=(float)((2*l+1)*wig_d(l,m,n,dh_beta(B1,b)));
    return;
  }
  e-=SZ_D1W;
  if (e < SZ_D12W){ // quadrature-weighted d^l on b1 grid [6,20,11,11]
    int l=(int)(e/(NB1*M2*M2)); int r=(int)(e%(NB1*M2*M2)); int b=r/(M2*M2); int mn=r%(M2*M2);
    int m=mn/M2-(B2-1), n=mn%M2-(B2-1);
    ws[OFF_D12W+e]=(float)(dh_w(B1,b)*wig_d(l,m,n,dh_beta(B1,b)));
    return;
  }
  e-=SZ_D12W;
  if (e < SZ_D2){ // conj Wigner-D at the 144 SO3 grid pts [6,144,11,11] complex
    int l=(int)(e/(144*M2*M2)); int r=(int)(e%(144*M2*M2)); int g=r/(M2*M2); int mn=r%(M2*M2);
    int m=mn/M2-(B2-1), n=mn%M2-(B2-1);
    int bi=g/48, ai=(g/6)%8, gi=g%6;
    double beta=(bi+1)*(PI_D/24.0), alpha=ai*(2.0*PI_D/8.0), gamma=gi*(2.0*PI_D/6.0);
    double d=wig_d(l,m,n,beta), ph=m*alpha+n*gamma;
    ws[OFF_D2R+e]=(float)( d*cos(ph));
    ws[OFF_D2I+e]=(float)(-d*sin(ph));
    return;
  }
  e-=SZ_D2;
  if (e < SZ_D2W){ // (2l+1)*d^l on b2 grid [6,12,11,11]
    int l=(int)(e/(NB2*M2*M2)); int r=(int)(e%(NB2*M2*M2)); int b=r/(M2*M2); int mn=r%(M2*M2);
    int m=mn/M2-(B2-1), n=mn%M2-(B2-1);
    ws[OFF_D2W+e]=(float)((2*l+1)*wig_d(l,m,n,dh_beta(B2,b)));
    return;
  }
  e-=SZ_D2W;
  if (e < SZ_W2N){ // normalized beta quadrature weights
    double s=0.0; for(int k=0;k<NB2;++k) s+=dh_w(B2,k);
    ws[OFF_W2N+e]=(float)(dh_w(B2,(int)e)/s);
    return;
  }
}

// ======================= stage 1: S2 azimuthal DFT + Legendre =======================
__global__ void k_s2fft(const float* __restrict__ x, float* __restrict__ ws){
  int t = blockIdx.x*blockDim.x+threadIdx.x;
  if (t >= BATCH*M1) return;
  int z=t/M1, mi=t%M1, m=mi-(B1-1);
  float axr[B1], axi[B1];
  #pragma unroll
  for(int l=0;l<B1;++l){ axr[l]=0.f; axi[l]=0.f; }
  const float* xz = x + (size_t)z*NB_IN*NB_IN;
  float w = -TWO_PI_F*(float)m/(float)NB_IN;
  for(int b=0;b<NB_IN;++b){
    float xr=0.f, xi=0.f;
    for(int a=0;a<NB_IN;++a){
      float s,c; __sincosf(w*(float)a,&s,&c);
      float v=xz[b*NB_IN+a];
      xr+=v*c; xi+=v*s;
    }
    #pragma unroll
    for(int l=0;l<B1;++l){
      float d=ws[OFF_WD + (size_t)l*NB_IN*M1 + (size_t)b*M1 + mi];
      axr[l]+=d*xr; axi[l]+=d*xi;
    }
  }
  #pragma unroll
  for(int l=0;l<B1;++l){
    ws[OFF_XR + (size_t)l*BATCH*M1 + t] = axr[l];
    ws[OFF_XI + (size_t)l*BATCH*M1 + t] = axi[l];
  }
}

// Psi1[l,o,n] = sum_g psi1[0,o,g]*Y1[l,g,n]
__global__ void k_psi1(const float* __restrict__ psi1, float* __restrict__ ws){
  int t=blockIdx.x*blockDim.x+threadIdx.x;
  if (t>=B1*F1*M1) return;
  int l=t/(F1*M1); int o=(t/M1)%F1; int ni=t%M1;
  float ar=0.f, ai=0.f;
  for(int g=0;g<24;++g){
    float p=psi1[o*24+g];
    ar+=p*ws[OFF_Y1R+(size_t)l*24*M1+(size_t)g*M1+ni];
    ai+=p*ws[OFF_Y1I+(size_t)l*24*M1+(size_t)g*M1+ni];
  }
  ws[OFF_P1R+t]=ar; ws[OFF_P1I+t]=ai;
}

// ======================= stage 2: fused Fh build + SO3 iFFT + ReLU =======================
__global__ void __launch_bounds__(128) k_y(float* __restrict__ ws){
  __shared__ float2 sF[M1*M1];
  __shared__ float2 sT[M1*NB1];
  int bid=blockIdx.x;
  int z=bid/(F1*NB1); int o=(bid/NB1)%F1; int b=bid%NB1;
  int tid=threadIdx.x;
  for(int idx=tid; idx<M1*M1; idx+=128){
    int mi=idx/M1, ni=idx%M1;
    float fr=0.f, fi=0.f;
    #pragma unroll
    for(int l=0;l<B1;++l){
      float xr=ws[OFF_XR+(size_t)l*BATCH*M1+(size_t)z*M1+mi];
      float xi=ws[OFF_XI+(size_t)l*BATCH*M1+(size_t)z*M1+mi];
      float pr=ws[OFF_P1R+(size_t)l*F1*M1+(size_t)o*M1+ni];
      float pi=ws[OFF_P1I+(size_t)l*F1*M1+(size_t)o*M1+ni];
      float d =ws[OFF_D1W+(size_t)l*NB1*M1*M1+(size_t)b*M1*M1+idx];
      fr += d*(xr*pr - xi*pi);
      fi += d*(xr*pi + xi*pr);
    }
    sF[idx]=make_float2(fr,fi);
  }
  __syncthreads();
  for(int idx=tid; idx<M1*NB1; idx+=128){
    int mi=idx/NB1, g=idx%NB1;
    float tr=0.f, ti=0.f;
    for(int ni=0;ni<M1;++ni){
      float ang = TWO_PI_F*(float)(ni-(B1-1))*(float)g/(float)NB1;
      float s,c; __sincosf(ang,&s,&c);
      float2 f=sF[mi*M1+ni];
      tr += f.x*c - f.y*s;
      ti += f.x*s + f.y*c;
    }
    sT[idx]=make_float2(tr,ti);
  }
  __syncthreads();
  float* yout = ws + OFF_Y + (size_t)bid*NB1*NB1;
  for(int idx=tid; idx<NB1*NB1; idx+=128){
    int a=idx/NB1, g=idx%NB1;
    float v=0.f;
    for(int mi=0;mi<M1;++mi){
      float ang = TWO_PI_F*(float)(mi-(B1-1))*(float)a/(float)NB1;
      float s,c; __sincosf(ang,&s,&c);
      float2 tc=sT[mi*NB1+g];
      v += tc.x*c - tc.y*s;
    }
    yout[idx] = v>0.f ? v : 0.f;
  }
}

// ======================= stage 3: SO3 forward FFT (20x20 -> 11x11) =======================
__global__ void __launch_bounds__(128) k_yf(float* __restrict__ ws){
  __shared__ float  sY[NB1*NB1];
  __shared__ float2 sT[M2*NB1];
  int bid=blockIdx.x;     // (z,i,b)
  int tid=threadIdx.x;
  const float* yin = ws + OFF_Y + (size_t)bid*NB1*NB1;
  for(int idx=tid; idx<NB1*NB1; idx+=128) sY[idx]=yin[idx];
  __syncthreads();
  for(int idx=tid; idx<M2*NB1; idx+=128){
    int mi=idx/NB1, g=idx%NB1;
    float tr=0.f, ti=0.f;
    for(int a=0;a<NB1;++a){
      float ang = -TWO_PI_F*(float)(mi-(B2-1))*(float)a/(float)NB1;
      float s,c; __sincosf(ang,&s,&c);
      float v=sY[a*NB1+g];
      tr+=v*c; ti+=v*s;
    }
    sT[idx]=make_float2(tr,ti);
  }
  __syncthreads();
  for(int idx=tid; idx<M2*M2; idx+=128){
    int mi=idx/M2, ni=idx%M2;
    float yr=0.f, yi=0.f;
    for(int g=0;g<NB1;++g){
      float ang = -TWO_PI_F*(float)(ni-(B2-1))*(float)g/(float)NB1;
      float s,c; __sincosf(ang,&s,&c);
      float2 tc=sT[mi*NB1+g];
      yr += tc.x*c - tc.y*s;
      yi += tc.x*s + tc.y*c;
    }
    ws[OFF_YFR+(size_t)bid*M2*M2+idx]=yr;
    ws[OFF_YFI+(size_t)bid*M2*M2+idx]=yi;
  }
}

// ======================= stage 4: beta contraction -> GEMM A operand =======================
__global__ void k_fx(float* __restrict__ ws){
  size_t t = blockIdx.x*(size_t)blockDim.x+threadIdx.x;
  constexpr size_t N = (size_t)B2*BATCH*F1*M2*M2;
  if (t >= N) return;
  int l = (int)(t/((size_t)BATCH*F1*M2*M2));
  size_t r = t%((size_t)BATCH*F1*M2*M2);
  int z = (int)(r/(F1*M2*M2));
  int i = (int)((r/(M2*M2))%F1);
  int mn = (int)(r%(M2*M2));
  int m=mn/M2, n=mn%M2;
  float ar=0.f, ai=0.f;
  for(int b=0;b<NB1;++b){
    float d=ws[OFF_D12W+(size_t)l*NB1*M2*M2+(size_t)b*M2*M2+mn];
    size_t yoff=((size_t)(z*F1+i)*NB1+b)*M2*M2+mn;
    ar+=d*ws[OFF_YFR+yoff];
    ai+=d*ws[OFF_YFI+yoff];
  }
  int row=z*M2+m, col=i*M2+n;
  ws[OFF_FXR+(size_t)l*GM*GK+(size_t)row*GK+col]=ar;
  ws[OFF_FXI+(size_t)l*GM*GK+(size_t)row*GK+col]=ai;
}

// Psi2 B operand: B[l][(i*11+k)][(o*11+n)] = sum_g psi2[i,o,g]*D2[l,g,k,n], zero-pad N->448
__global__ void k_psi2(const float* __restrict__ psi2, float* __restrict__ ws){
  size_t t=blockIdx.x*(size_t)blockDim.x+threadIdx.x;
  constexpr size_t N=(size_t)B2*GK*GNP;
  if (t >= N) return;
  int l=(int)(t/((size_t)GK*GNP));
  int rc=(int)(t%((size_t)GK*GNP));
  int row=rc/GNP, col=rc%GNP;
  float ar=0.f, ai=0.f;
  if (col < GN){
    int i=row/M2, k=row%M2;
    int o=col/M2, n=col%M2;
    for(int g=0;g<144;++g){
      float p=psi2[((size_t)i*F2+o)*144+g];
      size_t doff=(size_t)l*144*M2*M2+(size_t)g*M2*M2+(size_t)k*M2+n;
      ar+=p*ws[OFF_D2R+doff];
      ai+=p*ws[OFF_D2I+doff];
    }
  }
  ws[OFF_P2R+t]=ar; ws[OFF_P2I+t]=ai;
}

// ======================= stage 5: complex GEMM via V_WMMA_F32_16X16X4_F32 + TDM =======================
// Block = 1 tm x 4 tn tiles (4 waves, tm/l uniform per block).
// TDM (tensor_load_to_lds) DMAs the 16x220 complex A panel into LDS once per block;
// waves sync on TENSORcnt + workgroup barrier, then stream A from LDS (ds_load_b64)
// while B comes from global (L2-resident, 4.7 MB total).
// Complex product = 4 real WMMAs per K-step; -Ai*Bi via VALU negate (f32 WMMA NEG
// only supports C-negate per ISA 7.12).
constexpr int TM=GM/16;   // 88
constexpr int TN=GNP/16;  // 28
constexpr int TNG=TN/4;   // 7 groups of 4 n-tiles
constexpr unsigned LDS_A_HALF  = (unsigned)(16*GK*4);   // bytes per plane = 14080
constexpr unsigned LDS_A_BYTES = 2*LDS_A_HALF;          // 28160

// Build a 2D-tile D# (group0+group1) and issue TENSOR_LOAD_TO_LDS.
// dims/strides in 4-byte elements (data_size code 2).
__device__ __forceinline__ void tdm_load_2d(unsigned long long gaddr, unsigned lds_off,
                                            unsigned tensor_w, unsigned tensor_h,
                                            unsigned tile_w, unsigned tile_h,
                                            unsigned row_stride){
  u32x4 g0;
  g0[0] = 1u;                                                  // count=1 (user D#)
  g0[1] = lds_off;                                             // lds_addr (bytes)
  g0[2] = (unsigned)(gaddr & 0xffffffffull);                   // global_addr[31:0]
  g0[3] = (unsigned)((gaddr >> 32) & 0x01ffffffull) | (2u<<30);// global_addr[56:32] | type=2
  i32x8 g1;
  g1[0] = (int)(2u<<16);                                       // data_size=4B; no multicast
  g1[1] = (int)((tensor_w & 0xffffu) << 16);                   // tensor_dim0[15:0] @bits48
  g1[2] = (int)((tensor_w >> 16) | ((tensor_h & 0xffffu) << 16)); // dim0 hi | dim1 lo
  g1[3] = (int)((tensor_h >> 16) | ((tile_w & 0xffffu) << 16));   // dim1 hi | tile_dim0
  g1[4] = (int)(tile_h & 0xffffu);                             // tile_dim1 | tile_dim2=0
  g1[5] = (int)row_stride;                                     // tensor_dim0_stride[31:0]
  g1[6] = 0;                                                   // stride hi | dim1_stride lo
  g1[7] = 0;
  i32x4 z4 = {0,0,0,0};
#if __clang_major__ >= 23
  i32x8 z8 = {0,0,0,0,0,0,0,0};
  __builtin_amdgcn_tensor_load_to_lds(g0, g1, z4, z4, z8, 0);
#else
  __builtin_amdgcn_tensor_load_to_lds(g0, g1, z4, z4, 0);
#endif
}

__global__ void __launch_bounds__(128) k_gemm(float* __restrict__ ws){
  extern __shared__ float sA[];            // [2][16][GK] : r plane then i plane
  int wave = (int)(threadIdx.x>>5);
  int lane = (int)(threadIdx.x&31);
  int bid  = blockIdx.x;
  int l    = bid/(TM*TNG);
  int tm   = (bid/TNG)%TM;
  int tn   = (bid%TNG)*4 + wave;

  const float* Ar = ws+OFF_FXR+(size_t)l*GM*GK;
  const float* Ai = ws+OFF_FXI+(size_t)l*GM*GK;
  const float* Br = ws+OFF_P2R+(size_t)l*GK*GNP;
  const float* Bi = ws+OFF_P2I+(size_t)l*GK*GNP;

  // wave 0 DMAs the block's A panel (rows tm*16..+15, all K) into LDS via TDM
  if (wave == 0){
    unsigned long long gar = (unsigned long long)(uintptr_t)(Ar + (size_t)tm*16*GK);
    unsigned long long gai = (unsigned long long)(uintptr_t)(Ai + (size_t)tm*16*GK);
    tdm_load_2d(gar, 0u,         GK, GM, GK, 16, GK);
    tdm_load_2d(gai, LDS_A_HALF, GK, GM, GK, 16, GK);
    __builtin_amdgcn_s_wait_tensorcnt(0);
  }
  __syncthreads();

  __builtin_prefetch(Br, 0, 0);  // global_prefetch_b8: warm B panels (L2-resident)
  __builtin_prefetch(Bi, 0, 0);

  int rA16 = (lane&15)*GK;       // LDS row base for this lane's A row
  int colB = tn*16 + (lane&15);
  int khi  = (lane>>4)*2;
  v8f cr = {}; v8f ci = {};
  for(int k0=0;k0<GK;k0+=4){
    int kk=k0+khi;
    v2f ar = *reinterpret_cast<const v2f*>(&sA[rA16 + kk]);
    v2f ai = *reinterpret_cast<const v2f*>(&sA[16*GK + rA16 + kk]);
    v2f br, bi;
    br.x = Br[(size_t)kk*GNP+colB];     br.y = Br[(size_t)(kk+1)*GNP+colB];
    bi.x = Bi[(size_t)kk*GNP+colB];     bi.y = Bi[(size_t)(kk+1)*GNP+colB];
    v2f nai = -ai;
    cr = __builtin_amdgcn_wmma_f32_16x16x4_f32(false, ar,  false, br, (short)0, cr, false, false);
    cr = __builtin_amdgcn_wmma_f32_16x16x4_f32(false, nai, false, bi, (short)0, cr, false, false);
    ci = __builtin_amdgcn_wmma_f32_16x16x4_f32(false, ar,  false, bi, (short)0, ci, false, false);
    ci = __builtin_amdgcn_wmma_f32_16x16x4_f32(false, ai,  false, br, (short)0, ci, false, false);
  }
  float* For_ = ws+OFF_FOR+(size_t)l*GM*GNP;
  float* Foi_ = ws+OFF_FOI+(size_t)l*GM*GNP;
  int mbase = tm*16 + ((lane>>4)<<3);
  int col   = tn*16 + (lane&15);
  #pragma unroll
  for(int j=0;j<8;++j){
    For_[(size_t)(mbase+j)*GNP+col]=cr[j];
    Foi_[(size_t)(mbase+j)*GNP+col]=ci[j];
  }
}

// ======================= stage 6: fused Fh2 + iFFT + ReLU + quadrature integrate =======================
__global__ void k_zero(float* __restrict__ ws){
  int t=blockIdx.x*blockDim.x+threadIdx.x;
  if (t<(int)SZ_FEAT) ws[OFF_FEAT+t]=0.f;
}

__global__ void __launch_bounds__(128) k_y2feat(float* __restrict__ ws){
  __shared__ float2 sF[M2*M2];
  __shared__ float2 sT[M2*NB2];
  __shared__ float  sRed[128];
  int bid=blockIdx.x;
  int z=bid/(F2*NB2); int o=(bid/NB2)%F2; int b=bid%NB2;
  int tid=threadIdx.x;
  for(int idx=tid; idx<M2*M2; idx+=128){
    int m=idx/M2, n=idx%M2;
    float fr=0.f, fi=0.f;
    size_t pos=(size_t)(z*M2+m)*GNP + (size_t)(o*M2+n);
    #pragma unroll
    for(int l=0;l<B2;++l){
      float d=ws[OFF_D2W+(size_t)l*NB2*M2*M2+(size_t)b*M2*M2+idx];
      fr += d*ws[OFF_FOR+(size_t)l*GM*GNP+pos];
      fi += d*ws[OFF_FOI+(size_t)l*GM*GNP+pos];
    }
    sF[idx]=make_float2(fr,fi);
  }
  __syncthreads();
  for(int idx=tid; idx<M2*NB2; idx+=128){
    int mi=idx/NB2, g=idx%NB2;
    float tr=0.f, ti=0.f;
    for(int ni=0;ni<M2;++ni){
      float ang=TWO_PI_F*(float)(ni-(B2-1))*(float)g/(float)NB2;
      float s,c; __sincosf(ang,&s,&c);
      float2 f=sF[mi*M2+ni];
      tr+=f.x*c-f.y*s; ti+=f.x*s+f.y*c;
    }
    sT[idx]=make_float2(tr,ti);
  }
  __syncthreads();
  float part=0.f;
  for(int idx=tid; idx<NB2*NB2; idx+=128){
    int a=idx/NB2, g=idx%NB2;
    float v=0.f;
    for(int mi=0;mi<M2;++mi){
      float ang=TWO_PI_F*(float)(mi-(B2-1))*(float)a/(float)NB2;
      float s,c; __sincosf(ang,&s,&c);
      float2 tc=sT[mi*NB2+g];
      v += tc.x*c - tc.y*s;
    }
    part += v>0.f ? v : 0.f;
  }
  sRed[tid]=part;
  __syncthreads();
  for(int s=64;s>0;s>>=1){
    if (tid<s) sRed[tid]+=sRed[tid+s];
    __syncthreads();
  }
  if (tid==0){
    float w=ws[OFF_W2N+b];
    atomicAdd(ws+OFF_FEAT+(size_t)z*F2+o, sRed[0]*w/(float)(NB2*NB2));
  }
}

// ======================= stage 7: linear head =======================
__global__ void k_head(const float* __restrict__ wl, const float* __restrict__ bl,
                       const float* __restrict__ ws, float* __restrict__ out){
  int t=blockIdx.x*blockDim.x+threadIdx.x;
  if (t>=BATCH*FOUT) return;
  int z=t/FOUT, f=t%FOUT;
  float acc=bl[f];
  for(int o=0;o<F2;++o) acc += ws[OFF_FEAT+(size_t)z*F2+o]*wl[f*F2+o];
  out[t]=acc;
}

// ======================= launch =======================
extern "C" void kernel_launch(void* const* d_in, const int* in_sizes, int n_in,
                              void* d_out, int out_size, void* d_ws, size_t ws_size,
                              hipStream_t stream){
  (void)in_sizes; (void)n_in; (void)out_size; (void)ws_size;
  const float* x     = (const float*)d_in[0];
  const float* psi1  = (const float*)d_in[1];
  const float* psi2  = (const float*)d_in[2];
  const float* w_lin = (const float*)d_in[3];
  const float* b_lin = (const float*)d_in[4];
  float* ws  = (float*)d_ws;
  float* out = (float*)d_out;

  const size_t ntab = SZ_WD+SZ_Y1+SZ_D1W+SZ_D12W+SZ_D2+SZ_D2W+SZ_W2N;
  k_tables<<<(unsigned)((ntab+255)/256),256,0,stream>>>(ws);
  k_zero  <<<(unsigned)((SZ_FEAT+255)/256),256,0,stream>>>(ws);
  k_s2fft <<<(BATCH*M1+127)/128,128,0,stream>>>(x,ws);
  k_psi1  <<<(B1*F1*M1+127)/128,128,0,stream>>>(psi1,ws);
  k_y     <<<BATCH*F1*NB1,128,0,stream>>>(ws);
  k_yf    <<<BATCH*F1*NB1,128,0,stream>>>(ws);
  const size_t nfx=(size_t)B2*BATCH*F1*M2*M2;
  k_fx    <<<(unsigned)((nfx+255)/256),256,0,stream>>>(ws);
  const size_t np2=(size_t)B2*GK*GNP;
  k_psi2  <<<(unsigned)((np2+255)/256),256,0,stream>>>(psi2,ws);
  k_gemm  <<<B2*TM*TNG,128,LDS_A_BYTES,stream>>>(ws);
  k_y2feat<<<BATCH*F2*NB2,128,0,stream>>>(ws);
  k_head  <<<(BATCH*FOUT+255)/256,256,0,stream>>>(w_lin,b_lin,ws,out);
}